// GraphCoarsenLayer_65000035058039
// MI455X (gfx1250) — compile-verified
//
#include <hip/hip_runtime.h>
#include <hip/hip_bf16.h>

typedef float v2f __attribute__((ext_vector_type(2)));
typedef float v4f __attribute__((ext_vector_type(4)));
typedef float v8f __attribute__((ext_vector_type(8)));

#define CIN 256
#define OUTF 256
#define DNB 32
#define SNB 16
#define ROWS_PER_WG 32
#define THREADS 256
#define LDS_STRIDE 772  // 768 floats of [x|agg_s|agg_u] + 4 pad -> conflict-free phase-2 reads

__global__ __launch_bounds__(THREADS) void graph_coarsen_wmma(
    const float* __restrict__ x,
    const float* __restrict__ Wself,
    const float* __restrict__ bself,
    const float* __restrict__ Wneigh,
    const float* __restrict__ bneigh,
    const float* __restrict__ Wcoars,
    const float* __restrict__ bcoars,
    const int*   __restrict__ neighbors,
    float*       __restrict__ out,
    int Ntot)
{
    __shared__ float lds[ROWS_PER_WG * LDS_STRIDE];

    const int tid  = threadIdx.x;
    const int row0 = blockIdx.x * ROWS_PER_WG;

    // ---------------- Phase 1: gather + mean-pool into LDS (b128 loads) ----
    // 64 threads cover 256 columns (4 each); 4 row-groups of 8 rows.
    {
        const int t  = tid & 63;
        const int c  = t * 4;
        const int rh = tid >> 6;            // 0..3
        const float inv = 1.0f / (float)SNB;
        for (int r = 0; r < 8; ++r) {
            const int lr = rh * 8 + r;      // lds row 0..31
            int row = row0 + lr;
            if (row >= Ntot) row = Ntot - 1;   // clamp (uniform per row-group)
            float* ldr = &lds[lr * LDS_STRIDE];

            v4f xv = *(const v4f*)(x + (size_t)row * CIN + c);
            *(v4f*)(ldr + c) = xv;

            v4f accs = {0.f, 0.f, 0.f, 0.f};
            v4f accu = {0.f, 0.f, 0.f, 0.f};
            const int* nbr = neighbors + (size_t)row * DNB;
            #pragma unroll
            for (int s = 0; s < DNB; ++s) {
                const int nb = nbr[s];
                v4f v = *(const v4f*)(x + (size_t)nb * CIN + c);
                if (s < SNB) accs += v; else accu += v;
            }
            *(v4f*)(ldr + CIN     + c) = accs * inv;
            *(v4f*)(ldr + 2 * CIN + c) = accu * inv;
        }
    }
    __syncthreads();

    // ---------------- Phase 2: WMMA f32 16x16x4, K = 3*256 -----------------
    // Wave w owns N-tiles {w*32, w*32+16} and M-tiles {rows 0-15, 16-31}.
    // Each B fragment is reused across both M-tiles (halves weight traffic).
    const int lane = tid & 31;
    const int wave = tid >> 5;            // 0..7
    const int half = lane >> 4;           // 0: lanes 0-15, 1: lanes 16-31
    const int l    = lane & 15;

    v8f acc[2][2];                        // [mt][nt]
    #pragma unroll
    for (int mt = 0; mt < 2; ++mt)
        #pragma unroll
        for (int nt = 0; nt < 2; ++nt) acc[mt][nt] = (v8f){};

    const float* Ws[3] = { Wself, Wneigh, Wcoars };
    const int nbase = wave * 32;

    for (int m = 0; m < 3; ++m) {
        const float* __restrict__ W = Ws[m];
        // A sources: M-tile 0 -> lds row l, M-tile 1 -> lds row l+16
        const float* arow0 = &lds[l * LDS_STRIDE        + m * CIN + 2 * half];
        const float* arow1 = arow0 + 16 * LDS_STRIDE;
        #pragma unroll 4
        for (int k0 = 0; k0 < CIN; k0 += 4) {
            // A: VGPR v holds A[M][K = k0 + 2*half + v]  (contiguous -> ds_load_b64)
            v2f a0 = *(const v2f*)(arow0 + k0);
            v2f a1 = *(const v2f*)(arow1 + k0);
            const float* wk = W + (size_t)(k0 + 2 * half) * OUTF;
            #pragma unroll
            for (int nt = 0; nt < 2; ++nt) {
                const int n = nbase + nt * 16 + l;
                // B: VGPR v holds B[K = k0 + 2*half + v][N = n]
                v2f b;
                b.x = wk[n];
                b.y = wk[OUTF + n];
                acc[0][nt] = __builtin_amdgcn_wmma_f32_16x16x4_f32(
                    false, a0, false, b, (short)0, acc[0][nt], false, false);
                acc[1][nt] = __builtin_amdgcn_wmma_f32_16x16x4_f32(
                    false, a1, false, b, (short)0, acc[1][nt], false, false);
            }
        }
    }

    // ---------------- Epilogue: bias + store --------------------------------
    #pragma unroll
    for (int nt = 0; nt < 2; ++nt) {
        const int n = nbase + nt * 16 + l;
        const float bias = bself[n] + bneigh[n] + bcoars[n];
        #pragma unroll
        for (int mt = 0; mt < 2; ++mt) {
            v8f d = acc[mt][nt];
            #pragma unroll
            for (int r = 0; r < 8; ++r) {
                const int row = row0 + mt * 16 + r + 8 * half;  // D VGPR r: M = r + 8*half
                if (row < Ntot)
                    out[(size_t)row * OUTF + n] = d[r] + bias;
            }
        }
    }
}

extern "C" void kernel_launch(void* const* d_in, const int* in_sizes, int n_in,
                              void* d_out, int out_size, void* d_ws, size_t ws_size,
                              hipStream_t stream) {
    const float* x      = (const float*)d_in[0];
    const float* Wself  = (const float*)d_in[1];
    const float* bself  = (const float*)d_in[2];
    const float* Wneigh = (const float*)d_in[3];
    const float* bneigh = (const float*)d_in[4];
    const float* Wcoars = (const float*)d_in[5];
    const float* bcoars = (const float*)d_in[6];
    const int*   neigh  = (const int*)d_in[7];
    float* out = (float*)d_out;

    const int Ntot = in_sizes[0] / CIN;                       // 50000
    const int grid = (Ntot + ROWS_PER_WG - 1) / ROWS_PER_WG;  // 1563

    graph_coarsen_wmma<<<grid, THREADS, 0, stream>>>(
        x, Wself, bself, Wneigh, bneigh, Wcoars, bcoars, neigh, out, Ntot);
}